// UltimatePhysicsLoss_79671643341225
// MI455X (gfx1250) — compile-verified
//
#include <hip/hip_runtime.h>
#include <math.h>

// ---------------------------------------------------------------------------
// UltimatePhysicsLoss for MI455X (gfx1250, wave32).
//
// Core idea: every pairwise-distance matrix tile d2[16][16] is computed as a
// single V_WMMA_F32_16X16X4_F32:
//     A = -2 * rows (16x4, K padded 3->4 with 0)
//     B =      cols (4x16)
//     C[m][n] = |row_m|^2 + |col_n|^2        (folded into accumulator)
//     D = A x B + C = squared distances
// min / relu-sum reductions stay in VALU; sqrt is deferred to per-row minima
// (monotone), so only O(N) sqrts instead of O(N^2) for chamfer/penetration.
// ---------------------------------------------------------------------------

typedef __attribute__((ext_vector_type(2))) float v2f;
typedef __attribute__((ext_vector_type(8))) float v8f;

#if defined(__AMDGCN__) && __has_builtin(__builtin_amdgcn_wmma_f32_16x16x4_f32)
#define USE_WMMA_F32K4 1
#else
#define USE_WMMA_F32K4 0
#endif

constexpr int kB  = 16;
constexpr int kNH = 1024;
constexpr int kNO = 2048;
constexpr int kDQ = 24;
constexpr float kAlphaFocal = 0.25f;
constexpr float kMarginPen  = 0.005f;
constexpr float kRadiusRep  = 0.015f;

// ---- workspace layout (floats); total ~132K floats (~528 KB) ---------------
constexpr int OFF_MINH_PG = 0;                              // [B][NH]
constexpr int OFF_MINH_GP = OFF_MINH_PG + kB * kNH;         // [B][NH]
constexpr int OFF_MINO_PG = OFF_MINH_GP + kB * kNH;         // [B][NO]
constexpr int OFF_MINO_GP = OFF_MINO_PG + kB * kNO;         // [B][NO]
constexpr int OFF_MINPEN  = OFF_MINO_GP + kB * kNO;         // [B][NO]
constexpr int OFF_REPPART = OFF_MINPEN  + kB * kNO;         // [B][NO/16]
constexpr int OFF_BATCH4  = OFF_REPPART + kB * (kNO / 16);  // [B][4]
constexpr int OFF_BPOSE   = OFF_BATCH4  + kB * 4;           // [B]
constexpr int OFF_SCAL    = OFF_BPOSE   + kB;               // [2]: qpos, contact

// ---------------------------------------------------------------------------
// Per-wave 16-row strip: min over ALL columns of squared distance.
// One wave (32 lanes) per block; grid = B * (Nr/16).
// Row minima are wave-private -> no atomics, fully deterministic.
// ---------------------------------------------------------------------------
__global__ __launch_bounds__(32) void rowmin_d2_kernel(
    const float* __restrict__ rows, const float* __restrict__ cols,
    float* __restrict__ out_min, int Nr, int Nc, int rowTiles)
{
  const int wid  = blockIdx.x;
  const int b    = wid / rowTiles;
  const int rt   = wid - b * rowTiles;
  const int lane = threadIdx.x;
  const int li   = lane & 15;
  const bool hi  = lane >= 16;

  const float* rb = rows + (size_t)b * Nr * 3;
  const float* cb = cols + (size_t)b * Nc * 3;

  const int ridx = rt * 16 + li;
  const float rx = rb[ridx * 3 + 0];
  const float ry = rb[ridx * 3 + 1];
  const float rz = rb[ridx * 3 + 2];
  const float na = rx * rx + ry * ry + rz * rz;

#if USE_WMMA_F32K4
  // Row-norm broadcast: VGPR v of the C/D tile is row v (lanes<16) or v+8.
  float nav[8];
#pragma unroll
  for (int v = 0; v < 8; ++v) nav[v] = __shfl(na, hi ? (v + 8) : v, 32);
  // A operand (16x4 f32): lanes 0-15 = (x,y) of row li; lanes 16-31 = (z,0).
  // Pre-scaled by -2 so the WMMA directly produces -2*a.b.
  v2f aop;
  aop[0] = hi ? (-2.f * rz) : (-2.f * rx);
  aop[1] = hi ? 0.f        : (-2.f * ry);
#else
  float rxv[8], ryv[8], rzv[8];
#pragma unroll
  for (int v = 0; v < 8; ++v) {
    const int src = hi ? (v + 8) : v;
    rxv[v] = __shfl(rx, src, 32);
    ryv[v] = __shfl(ry, src, 32);
    rzv[v] = __shfl(rz, src, 32);
  }
#endif

  v8f acc;
#pragma unroll
  for (int v = 0; v < 8; ++v) acc[v] = 3.0e38f;

  const int cTiles = Nc >> 4;
  for (int ct = 0; ct < cTiles; ++ct) {
    const int cidx = ct * 16 + li;
    const float cx = cb[cidx * 3 + 0];
    const float cy = cb[cidx * 3 + 1];
    const float cz = cb[cidx * 3 + 2];
#if USE_WMMA_F32K4
    const float nb = cx * cx + cy * cy + cz * cz;
    // B operand (4x16 f32): mirrored layout of A -> identical load pattern.
    v2f bop;
    bop[0] = hi ? cz  : cx;
    bop[1] = hi ? 0.f : cy;
    // Accumulator carries |a|^2 + |b|^2; the WMMA adds -2*a.b -> d^2.
    v8f c;
#pragma unroll
    for (int v = 0; v < 8; ++v) c[v] = nav[v] + nb;
    v8f d = __builtin_amdgcn_wmma_f32_16x16x4_f32(
        false, aop, false, bop, (short)0, c, false, false);
#pragma unroll
    for (int v = 0; v < 8; ++v) acc[v] = fminf(acc[v], d[v]);
#else
#pragma unroll
    for (int v = 0; v < 8; ++v) {
      const float dx = rxv[v] - cx, dy = ryv[v] - cy, dz = rzv[v] - cz;
      acc[v] = fminf(acc[v], dx * dx + dy * dy + dz * dz);
    }
#endif
  }

  // Min across the 16 lanes holding each row (xor-shuffles stay in each half).
#pragma unroll
  for (int m = 1; m < 16; m <<= 1) {
#pragma unroll
    for (int v = 0; v < 8; ++v) acc[v] = fminf(acc[v], __shfl_xor(acc[v], m, 32));
  }
  if (li == 0) {
    const int base = b * Nr + rt * 16 + (hi ? 8 : 0);
#pragma unroll
    for (int v = 0; v < 8; ++v) out_min[base + v] = acc[v];
  }
}

// ---------------------------------------------------------------------------
// Repulsion: sum over all pairs of relu(R - dist), diagonal excluded.
// Same tiling; each wave writes one deterministic partial sum (no atomics).
// ---------------------------------------------------------------------------
__global__ __launch_bounds__(32) void repulsion_kernel(
    const float* __restrict__ pts, float* __restrict__ rep_part,
    int N, int rowTiles)
{
  const int wid  = blockIdx.x;
  const int b    = wid / rowTiles;
  const int rt   = wid - b * rowTiles;
  const int lane = threadIdx.x;
  const int li   = lane & 15;
  const bool hi  = lane >= 16;

  const float* pb = pts + (size_t)b * N * 3;
  const int ridx = rt * 16 + li;
  const float rx = pb[ridx * 3 + 0];
  const float ry = pb[ridx * 3 + 1];
  const float rz = pb[ridx * 3 + 2];
  const float na = rx * rx + ry * ry + rz * rz;

#if USE_WMMA_F32K4
  float nav[8];
#pragma unroll
  for (int v = 0; v < 8; ++v) nav[v] = __shfl(na, hi ? (v + 8) : v, 32);
  v2f aop;
  aop[0] = hi ? (-2.f * rz) : (-2.f * rx);
  aop[1] = hi ? 0.f        : (-2.f * ry);
#else
  float rxv[8], ryv[8], rzv[8];
#pragma unroll
  for (int v = 0; v < 8; ++v) {
    const int src = hi ? (v + 8) : v;
    rxv[v] = __shfl(rx, src, 32);
    ryv[v] = __shfl(ry, src, 32);
    rzv[v] = __shfl(rz, src, 32);
  }
#endif

  float s = 0.f;
  const int rowBase = rt * 16 + (hi ? 8 : 0);
  const int cTiles = N >> 4;
  for (int ct = 0; ct < cTiles; ++ct) {
    const int cidx = ct * 16 + li;
    const float cx = pb[cidx * 3 + 0];
    const float cy = pb[cidx * 3 + 1];
    const float cz = pb[cidx * 3 + 2];
#if USE_WMMA_F32K4
    const float nb = cx * cx + cy * cy + cz * cz;
    v2f bop;
    bop[0] = hi ? cz  : cx;
    bop[1] = hi ? 0.f : cy;
    v8f c;
#pragma unroll
    for (int v = 0; v < 8; ++v) c[v] = nav[v] + nb;
    v8f d2v = __builtin_amdgcn_wmma_f32_16x16x4_f32(
        false, aop, false, bop, (short)0, c, false, false);
#endif
#pragma unroll
    for (int v = 0; v < 8; ++v) {
#if USE_WMMA_F32K4
      float d2 = d2v[v];
#else
      const float dx = rxv[v] - cx, dy = ryv[v] - cy, dz = rzv[v] - cz;
      float d2 = dx * dx + dy * dy + dz * dz;
#endif
      d2 = fmaxf(d2, 1e-12f);
      float r = fmaxf(kRadiusRep - sqrtf(d2), 0.f);
      if (rowBase + v == cidx) r = 0.f;  // diagonal -> inf -> relu 0
      s += r;
    }
  }
#pragma unroll
  for (int m = 1; m < 32; m <<= 1) s += __shfl_xor(s, m, 32);
  if (lane == 0) rep_part[b * rowTiles + rt] = s;
}

// ---------------------------------------------------------------------------
__device__ __forceinline__ float block_sum(float v, float* sm) {
  const int t = threadIdx.x;
  sm[t] = v;
  __syncthreads();
  for (int s = blockDim.x >> 1; s > 0; s >>= 1) {
    if (t < s) sm[t] += sm[t + s];
    __syncthreads();
  }
  const float r = sm[0];
  __syncthreads();
  return r;
}

// Focal + qpos MSE + per-batch pose loss. One block.
__global__ __launch_bounds__(256) void small_losses_kernel(
    const float* __restrict__ logits, const float* __restrict__ gt_contact,
    const float* __restrict__ pred_qpos, const float* __restrict__ gt_qpos,
    const float* __restrict__ pred_pose, const float* __restrict__ gt_pose,
    float* __restrict__ scal, float* __restrict__ batch_pose)
{
  __shared__ float sm[256];
  const int t = threadIdx.x;

  float sf = 0.f;
  for (int i = t; i < kB * kNH; i += 256) {
    const float x = logits[i];
    const float y = gt_contact[i];
    const float bce = fmaxf(x, 0.f) - x * y + log1pf(expf(-fabsf(x)));
    const float pt = expf(-bce);
    const float om = 1.f - pt;
    sf += kAlphaFocal * om * om * bce;
  }
  sf = block_sum(sf, sm);

  float sq = 0.f;
  for (int i = t; i < kB * kDQ; i += 256) {
    const float d = pred_qpos[i] - gt_qpos[i];
    sq += d * d;
  }
  sq = block_sum(sq, sm);

  if (t < kB) {
    const int b = t;
    const float* pp = pred_pose + b * 7;
    const float* gp = gt_pose + b * 7;
    const float lt = (fabsf(pp[0] - gp[0]) + fabsf(pp[1] - gp[1]) +
                      fabsf(pp[2] - gp[2])) * (1.f / 3.f);
    float npn = 0.f, ngn = 0.f, dq = 0.f;
#pragma unroll
    for (int i = 0; i < 4; ++i) {
      npn += pp[3 + i] * pp[3 + i];
      ngn += gp[3 + i] * gp[3 + i];
      dq  += pp[3 + i] * gp[3 + i];
    }
    dq *= rsqrtf(npn) * rsqrtf(ngn);
    batch_pose[b] = lt + 0.1f * (1.f - fabsf(dq));
  }
  if (t == 0) {
    scal[0] = sq / (float)(kB * kDQ);   // loss_qpos
    scal[1] = sf / (float)(kB * kNH);   // loss_contact
  }
}

// Per-batch: sqrt the row minima, means, repulsion/pen normalization.
__global__ __launch_bounds__(256) void batch_reduce_kernel(
    const float* __restrict__ minH_pg, const float* __restrict__ minH_gp,
    const float* __restrict__ minO_pg, const float* __restrict__ minO_gp,
    const float* __restrict__ minPen,  const float* __restrict__ rep_part,
    float* __restrict__ batch4)
{
  __shared__ float sm[256];
  const int b = blockIdx.x;
  const int t = threadIdx.x;

  float sh = 0.f;
  for (int i = t; i < kNH; i += 256) {
    sh += sqrtf(fmaxf(minH_pg[b * kNH + i], 1e-12f));
    sh += sqrtf(fmaxf(minH_gp[b * kNH + i], 1e-12f));
  }
  sh = block_sum(sh, sm);

  float so = 0.f;
  for (int i = t; i < kNO; i += 256) {
    so += sqrtf(fmaxf(minO_pg[b * kNO + i], 1e-12f));
    so += sqrtf(fmaxf(minO_gp[b * kNO + i], 1e-12f));
  }
  so = block_sum(so, sm);

  float sp = 0.f;
  for (int i = t; i < kNO; i += 256) {
    sp += fmaxf(kMarginPen - sqrtf(fmaxf(minPen[b * kNO + i], 1e-12f)), 0.f);
  }
  sp = block_sum(sp, sm);

  float sr = 0.f;
  for (int i = t; i < kNO / 16; i += 256) sr += rep_part[b * (kNO / 16) + i];
  sr = block_sum(sr, sm);

  if (t == 0) {
    batch4[b * 4 + 0] = sh * (1.f / kNH);                       // chamfer hand
    batch4[b * 4 + 1] = so * (1.f / kNO);                       // chamfer obj
    batch4[b * 4 + 2] = sr * (1.f / ((float)kNO * (float)kNO)); // repulsion
    batch4[b * 4 + 3] = sp * (1.f / kNO);                       // penetration
  }
}

__global__ __launch_bounds__(32) void finalize_kernel(
    const float* __restrict__ batch4, const float* __restrict__ batch_pose,
    const float* __restrict__ scal, const float* __restrict__ has_contact,
    float* __restrict__ out)
{
  if (threadIdx.x != 0 || blockIdx.x != 0) return;
  float valid = 1e-6f;
  for (int b = 0; b < kB; ++b) valid += has_contact[b];
  float cdh = 0.f, cdo = 0.f, pos = 0.f, rep = 0.f, pen = 0.f;
  for (int b = 0; b < kB; ++b) {
    const float h = has_contact[b];
    cdh += batch4[b * 4 + 0];
    cdo += batch4[b * 4 + 1] * h;
    rep += batch4[b * 4 + 2] * h;
    pen += batch4[b * 4 + 3] * h;
    pos += batch_pose[b] * h;
  }
  cdh *= (1.f / kB);
  const float inv = 1.f / valid;
  out[0] = 5.f * cdh + 5.f * cdo * inv + 2.f * pos * inv + 1.f * scal[0] +
           2.f * scal[1] + 0.5f * rep * inv + 0.5f * pen * inv;
}

// ---------------------------------------------------------------------------
extern "C" void kernel_launch(void* const* d_in, const int* in_sizes, int n_in,
                              void* d_out, int out_size, void* d_ws, size_t ws_size,
                              hipStream_t stream) {
  (void)in_sizes; (void)n_in; (void)out_size; (void)ws_size;
  const float* pred_hand   = (const float*)d_in[0];
  const float* pred_obj    = (const float*)d_in[1];
  const float* pred_pose   = (const float*)d_in[2];
  const float* pred_qpos   = (const float*)d_in[3];
  const float* pred_logit  = (const float*)d_in[4];
  const float* gt_hand     = (const float*)d_in[5];
  const float* gt_obj      = (const float*)d_in[6];
  const float* gt_pose     = (const float*)d_in[7];
  const float* gt_qpos     = (const float*)d_in[8];
  const float* gt_contact  = (const float*)d_in[9];
  const float* has_contact = (const float*)d_in[10];

  float* ws       = (float*)d_ws;
  float* out      = (float*)d_out;
  float* minH_pg  = ws + OFF_MINH_PG;
  float* minH_gp  = ws + OFF_MINH_GP;
  float* minO_pg  = ws + OFF_MINO_PG;
  float* minO_gp  = ws + OFF_MINO_GP;
  float* minPen   = ws + OFF_MINPEN;
  float* rep_part = ws + OFF_REPPART;
  float* batch4   = ws + OFF_BATCH4;
  float* bpose    = ws + OFF_BPOSE;
  float* scal     = ws + OFF_SCAL;

  // Chamfer hand, both directions (rows own their minima -> exact, no init).
  rowmin_d2_kernel<<<kB * (kNH / 16), 32, 0, stream>>>(
      pred_hand, gt_hand, minH_pg, kNH, kNH, kNH / 16);
  rowmin_d2_kernel<<<kB * (kNH / 16), 32, 0, stream>>>(
      gt_hand, pred_hand, minH_gp, kNH, kNH, kNH / 16);
  // Chamfer obj, both directions.
  rowmin_d2_kernel<<<kB * (kNO / 16), 32, 0, stream>>>(
      pred_obj, gt_obj, minO_pg, kNO, kNO, kNO / 16);
  rowmin_d2_kernel<<<kB * (kNO / 16), 32, 0, stream>>>(
      gt_obj, pred_obj, minO_gp, kNO, kNO, kNO / 16);
  // Penetration: per obj point, min over hand points.
  rowmin_d2_kernel<<<kB * (kNO / 16), 32, 0, stream>>>(
      pred_obj, pred_hand, minPen, kNO, kNH, kNO / 16);
  // Repulsion partials.
  repulsion_kernel<<<kB * (kNO / 16), 32, 0, stream>>>(
      pred_obj, rep_part, kNO, kNO / 16);
  // Small losses + per-batch pose.
  small_losses_kernel<<<1, 256, 0, stream>>>(
      pred_logit, gt_contact, pred_qpos, gt_qpos, pred_pose, gt_pose, scal, bpose);
  // Per-batch means.
  batch_reduce_kernel<<<kB, 256, 0, stream>>>(
      minH_pg, minH_gp, minO_pg, minO_gp, minPen, rep_part, batch4);
  // Weighted combine.
  finalize_kernel<<<1, 32, 0, stream>>>(batch4, bpose, scal, has_contact, out);
}